// NTXent_28097676050918
// MI455X (gfx1250) — compile-verified
//
#include <hip/hip_runtime.h>
#include <hip/hip_bf16.h>

typedef __attribute__((ext_vector_type(2))) float v2f;
typedef __attribute__((ext_vector_type(8))) float v8f;

#define N_ROWS   4096
#define TWO_N    8192
#define DIM      128
#define NT       (TWO_N / 16)            // 512 column tiles
#define INV_T    2.0f                    // 1 / temperature
#define LOG2E    1.4426950408889634f
#define LN2      0.6931471805599453f
#define LDS_STRIDE 132                   // 128 + 4 pad -> conflict-free v2f reads

// ---------------------------------------------------------------------------
// Kernel 1: row-wise L2 normalize, concat [emb_i; emb_j] -> reps [8192 x 128]
// One wave per row, float4 per lane (32 lanes * 4 = 128).
// ---------------------------------------------------------------------------
__global__ __launch_bounds__(256) void ntxent_normalize(
    const float* __restrict__ emb_i, const float* __restrict__ emb_j,
    float* __restrict__ reps) {
  const int row  = (int)((blockIdx.x * 256 + threadIdx.x) >> 5);
  const int lane = threadIdx.x & 31;
  if (row >= TWO_N) return;
  const float* src = (row < N_ROWS) ? (emb_i + (size_t)row * DIM)
                                    : (emb_j + (size_t)(row - N_ROWS) * DIM);
  float4 v = reinterpret_cast<const float4*>(src)[lane];
  float ss = v.x * v.x + v.y * v.y + v.z * v.z + v.w * v.w;
  ss += __shfl_xor(ss, 16, 32);
  ss += __shfl_xor(ss, 8, 32);
  ss += __shfl_xor(ss, 4, 32);
  ss += __shfl_xor(ss, 2, 32);
  ss += __shfl_xor(ss, 1, 32);
  const float inv = 1.0f / fmaxf(__builtin_sqrtf(ss), 1e-12f);
  v.x *= inv; v.y *= inv; v.z *= inv; v.w *= inv;
  reinterpret_cast<float4*>(reps + (size_t)row * DIM)[lane] = v;
}

// ---------------------------------------------------------------------------
// Kernel 2: fused sim-GEMM + masked exp-sum + positive extraction.
// 64 blocks x 256 threads (8 waves). Wave w owns rows [blk*128 + 16w, +16).
// A (16x128) preloaded to VGPRs once. B strips double-buffered through LDS
// with register-staged global loads: issue next tile's global_load_b128,
// run 32 back-to-back f32 WMMAs + exp folding over the current tile (hiding
// L2 latency), then ds_store the staged tile, one barrier per iteration.
// f32 WMMA layout (16x16x4): lane = {h,r}, r=lane&15 (=M or N), h=lane>>4
//   selects K pair; v2f = {K=2h, K=2h+1}. C/D: element (m,n) is in
//   lane n + 16*(m>=8), vgpr m&7.
// ---------------------------------------------------------------------------
__global__ __launch_bounds__(256) void ntxent_main(
    const float* __restrict__ reps, float* __restrict__ partials) {
  __shared__ float ldsB[2][16 * LDS_STRIDE];

  const int lane = threadIdx.x & 31;
  const int wid  = threadIdx.x >> 5;         // 0..7
  const int r    = lane & 15;
  const int h    = lane >> 4;
  const int rowbase = (int)blockIdx.x * 128 + wid * 16;
  const int posbase = (rowbase + N_ROWS) & (TWO_N - 1);

  // This thread's staging slots: rows {wid, wid+8} of the 16-row strip,
  // float4 column = lane.
  const int strow0 = wid;
  const int strow1 = wid + 8;
  const int stcol  = lane * 4;

  // Preload this wave's A strip (16 rows x 128) into 64 VGPRs.
  v2f a[32];
  {
    const float* arow = reps + (size_t)(rowbase + r) * DIM + 2 * h;
#pragma unroll
    for (int k = 0; k < 32; ++k)
      a[k] = *reinterpret_cast<const v2f*>(arow + 4 * k);
  }

  float denomAcc[8];
#pragma unroll
  for (int v = 0; v < 8; ++v) denomAcc[v] = 0.0f;
  float posAcc = 0.0f;

  // Prologue: stage tile 0 into buffer 0.
  {
    const float4 p0 = *reinterpret_cast<const float4*>(
        reps + (size_t)strow0 * DIM + stcol);
    const float4 p1 = *reinterpret_cast<const float4*>(
        reps + (size_t)strow1 * DIM + stcol);
    *reinterpret_cast<float4*>(&ldsB[0][strow0 * LDS_STRIDE + stcol]) = p0;
    *reinterpret_cast<float4*>(&ldsB[0][strow1 * LDS_STRIDE + stcol]) = p1;
  }
  __syncthreads();

  for (int ct = 0; ct < NT; ++ct) {
    const int  cur      = ct & 1;
    const int  colbase  = ct * 16;
    const bool havenext = (ct + 1 < NT);

    // (1) Issue next tile's global loads into registers (LOADcnt in flight
    //     across the whole compute section below).
    float4 s0 = {}, s1 = {};
    if (havenext) {
      const float* g = reps + (size_t)(colbase + 16) * DIM;
      s0 = *reinterpret_cast<const float4*>(g + (size_t)strow0 * DIM + stcol);
      s1 = *reinterpret_cast<const float4*>(g + (size_t)strow1 * DIM + stcol);
    }

    // (2) Pull all 32 B operands for this tile into registers first
    //     (16x ds_load_2addr_b64, one DScnt drain), ...
    v2f b[32];
    const float* browp = &ldsB[cur][r * LDS_STRIDE + 2 * h];
#pragma unroll
    for (int k = 0; k < 32; ++k)
      b[k] = *reinterpret_cast<const v2f*>(browp + 4 * k);

    // (3) ... then 32 back-to-back K=4 f32 WMMAs (exact precision).
    v8f c = {};
#pragma unroll
    for (int k = 0; k < 32; ++k)
      c = __builtin_amdgcn_wmma_f32_16x16x4_f32(false, a[k], false, b[k],
                                                (short)0, c, false, false);

    // (4) Fold tile into per-row exp-sums; diagonals are tile-aligned
    //     (16 | 4096), positives sit on the tile diagonal of one col tile.
    const bool diagT = (colbase == rowbase);
    const bool posT  = (colbase == posbase);
    const int  m0    = h << 3;
#pragma unroll
    for (int v = 0; v < 8; ++v) {
      const float s   = c[v] * INV_T;        // sim / T
      const bool  hit = ((m0 + v) == r);     // tile-diagonal element
      if (posT && hit) posAcc += s;
      const float e = __builtin_amdgcn_exp2f(s * LOG2E);
      if (!(diagT && hit)) denomAcc[v] += e;
    }

    // (5) Commit the staged tile to the other buffer and synchronize.
    if (havenext) {
      float* dbuf = &ldsB[cur ^ 1][0];
      *reinterpret_cast<float4*>(&dbuf[strow0 * LDS_STRIDE + stcol]) = s0;
      *reinterpret_cast<float4*>(&dbuf[strow1 * LDS_STRIDE + stcol]) = s1;
    }
    __syncthreads();
  }

  // Reduce denominators across the 16 lanes holding each row.
#pragma unroll
  for (int v = 0; v < 8; ++v) {
    float d = denomAcc[v];
    d += __shfl_xor(d, 1, 32);
    d += __shfl_xor(d, 2, 32);
    d += __shfl_xor(d, 4, 32);
    d += __shfl_xor(d, 8, 32);
    denomAcc[v] = d;  // lanes 0 and 16 now hold full sums for rows h*8+v
  }

  // loss_i = -pos_i/T + log(denom_i); sum this wave's 16 rows.
  float t = -posAcc;  // each positive counted in exactly one lane
  if (r == 0) {
#pragma unroll
    for (int v = 0; v < 8; ++v)
      t += __builtin_amdgcn_logf(denomAcc[v]) * LN2;  // v_log_f32 is log2
  }
  t += __shfl_xor(t, 1, 32);
  t += __shfl_xor(t, 2, 32);
  t += __shfl_xor(t, 4, 32);
  t += __shfl_xor(t, 8, 32);
  t += __shfl_xor(t, 16, 32);
  if (lane == 0) partials[(int)blockIdx.x * 8 + wid] = t;
}

// ---------------------------------------------------------------------------
// Kernel 3: deterministic reduction of 512 wave partials -> scalar loss.
// ---------------------------------------------------------------------------
__global__ __launch_bounds__(256) void ntxent_finalize(
    const float* __restrict__ partials, float* __restrict__ out, int n) {
  __shared__ float red[8];
  const int lane = threadIdx.x & 31;
  const int wid  = threadIdx.x >> 5;
  float v = 0.0f;
  for (int i = (int)threadIdx.x; i < n; i += 256) v += partials[i];
  v += __shfl_xor(v, 1, 32);
  v += __shfl_xor(v, 2, 32);
  v += __shfl_xor(v, 4, 32);
  v += __shfl_xor(v, 8, 32);
  v += __shfl_xor(v, 16, 32);
  if (lane == 0) red[wid] = v;
  __syncthreads();
  if (threadIdx.x == 0) {
    float t = 0.0f;
#pragma unroll
    for (int i = 0; i < 8; ++i) t += red[i];
    out[0] = t / (float)TWO_N;
  }
}

extern "C" void kernel_launch(void* const* d_in, const int* in_sizes, int n_in,
                              void* d_out, int out_size, void* d_ws, size_t ws_size,
                              hipStream_t stream) {
  (void)in_sizes; (void)n_in; (void)out_size; (void)ws_size;
  const float* emb_i = (const float*)d_in[0];
  const float* emb_j = (const float*)d_in[1];
  float* reps     = (float*)d_ws;                       // 8192*128 f32 = 4 MB
  float* partials = reps + (size_t)TWO_N * DIM;         // 512 f32
  float* out      = (float*)d_out;

  ntxent_normalize<<<TWO_N / 8, 256, 0, stream>>>(emb_i, emb_j, reps);
  ntxent_main<<<TWO_N / 128, 256, 0, stream>>>(reps, partials);
  ntxent_finalize<<<1, 256, 0, stream>>>(partials, out, (TWO_N / 128) * 8);
}